// LinearRecurrentCell_8564164788383
// MI455X (gfx1250) — compile-verified
//
#include <hip/hip_runtime.h>

// CDNA5 / gfx1250: wave32, WMMA 16x16x4 FP32.
typedef float v2f __attribute__((ext_vector_type(2)));
typedef float v4f __attribute__((ext_vector_type(4)));
typedef float v8f __attribute__((ext_vector_type(8)));

#define HID 128            // hidden dim
#define KSTEPS (HID / 4)   // 32 wmma k-steps
#define ROWS_PER_WAVE 16
#define WAVES_PER_BLOCK 8
#define ROWS_PER_BLOCK (ROWS_PER_WAVE * WAVES_PER_BLOCK) // 128

// LDS holds W in a k-chunked layout:
//   ldsW[(k>>2)*512 + g*4 + (k&3)]  (128*128 floats = 64 KB exactly)
// WMMA A-fragment read: lane L reads g = g0 + L%16, k = 4*ks + (L<16?0:2)
//   float index = ks*512 + g*4 + koff  -> banks {0,4,..,60} (low lanes) and
//   {2,6,..,62} (high lanes): fully conflict-free ds_load_b64.

__global__ __launch_bounds__(256)
void lrc_wmma_f32_kernel(const float* __restrict__ x,
                         const float* __restrict__ h,
                         const float* __restrict__ W,
                         float* __restrict__ out)
{
    __shared__ float ldsW[HID * HID]; // 64 KB (320 KB/WGP -> multiple blocks resident)

    const int tid = threadIdx.x;

    // ---- Stage W (row-major in global) into k-chunked LDS layout, once per block.
    #pragma unroll
    for (int i = 0; i < (HID * HID) / 256; ++i) {
        const int idx = i * 256 + tid;        // coalesced global read
        const int g = idx >> 7;               // row of W (output feature)
        const int k = idx & (HID - 1);        // col of W (input feature)
        ldsW[(k >> 2) * 512 + g * 4 + (k & 3)] = W[idx];
    }
    __syncthreads();

    const int wave = tid >> 5;                // wave32
    const int lane = tid & 31;
    const int lm   = lane & 15;               // matrix row/col within tile
    const int koff = (lane < 16) ? 0 : 2;     // A/B K-pair select per ISA layout
    const int moff = (lane < 16) ? 0 : 8;     // C/D row offset per ISA layout

    const int row = (blockIdx.x * WAVES_PER_BLOCK + wave) * ROWS_PER_WAVE + lm;
    const float* __restrict__ hrow = h   + (size_t)row * HID;
    const float* __restrict__ xrow = x   + (size_t)row * HID;
    float*       __restrict__ orow = out + (size_t)row * HID;

    // ---- Preload all h (B-operand) fragments for this wave's 16 batch rows.
    // B[kk][n] = h[b0+n][4*ks+kk]; lane holds n = lm, kk = koff/koff+1.
    // Regular-temporal: each 128B line is re-read by 8 nearby b64 loads (L0 reuse).
    v2f hb[KSTEPS];
    #pragma unroll
    for (int ks = 0; ks < KSTEPS; ++ks) {
        hb[ks] = *(const v2f*)(hrow + ks * 4 + koff); // 8B aligned
    }

    // ---- 8 output g-tiles of 16 columns each.
    #pragma unroll
    for (int t = 0; t < 8; ++t) {
        const int g0 = t * 16;

        // C init from x: lane's 8 accumulators are out columns g0+moff .. g0+moff+7.
        // x is a pure stream (read once) -> non-temporal load.
        const v4f xa = __builtin_nontemporal_load((const v4f*)(xrow + g0 + moff));
        const v4f xb = __builtin_nontemporal_load((const v4f*)(xrow + g0 + moff + 4));
        v8f c = { xa.x, xa.y, xa.z, xa.w, xb.x, xb.y, xb.z, xb.w };

        const float* __restrict__ wfrag = ldsW + (g0 + lm) * 4 + koff;

        #pragma unroll
        for (int ks = 0; ks < KSTEPS; ++ks) {
            // A[m][kk] = W[g0+m][4*ks+kk] from LDS, conflict-free b64 load.
            const v2f a = *(const v2f*)(wfrag + ks * 512);
            // D = A(16x4) x B(4x16) + C ; fp32 WMMA
            c = __builtin_amdgcn_wmma_f32_16x16x4_f32(
                    /*neg_a=*/false, a,
                    /*neg_b=*/false, hb[ks],
                    /*c_mod=*/(short)0, c,
                    /*reuse_a=*/false, /*reuse_b=*/false);
        }

        // out is write-once -> non-temporal store (don't churn 192MB L2).
        const v4f oa = { c[0], c[1], c[2], c[3] };
        const v4f ob = { c[4], c[5], c[6], c[7] };
        __builtin_nontemporal_store(oa, (v4f*)(orow + g0 + moff));
        __builtin_nontemporal_store(ob, (v4f*)(orow + g0 + moff + 4));
    }
}

extern "C" void kernel_launch(void* const* d_in, const int* in_sizes, int n_in,
                              void* d_out, int out_size, void* d_ws, size_t ws_size,
                              hipStream_t stream) {
    const float* x = (const float*)d_in[0];   // x_projected_t : (B, H) fp32
    const float* h = (const float*)d_in[1];   // h_prev        : (B, H) fp32
    const float* W = (const float*)d_in[2];   // W_hh          : (H, H) fp32
    float* out = (float*)d_out;               // h_next        : (B, H) fp32

    const long long Btot = (long long)in_sizes[0] / HID;  // 1048576 rows
    const int grid = (int)(Btot / ROWS_PER_BLOCK);        // 8192 blocks

    hipLaunchKernelGGL(lrc_wmma_f32_kernel, dim3(grid), dim3(256), 0, stream,
                       x, h, W, out);
}